// HCF_26585847562808
// MI455X (gfx1250) — compile-verified
//
#include <hip/hip_runtime.h>

// LightGCN propagation: chained COO SpMMs, D=128, mean over layers.
// Gather of h[col] rows done by the CDNA5 Tensor Data Mover in gather mode
// (8 indexed rows -> LDS per descriptor, double-buffered, TENSORcnt-tracked).
// Scatter done with global f32 atomics (L2-resident working set on MI455X).

#define DD 128
#define CHUNK 8                 // edges per TDM gather descriptor (32-bit idx mode max)
#define SPMM_WAVES 4            // waves per block
#define SPMM_BLOCK (SPMM_WAVES * 32)

typedef __attribute__((ext_vector_type(4))) unsigned int u32x4;
typedef __attribute__((ext_vector_type(8))) int          i32x8;
typedef __attribute__((ext_vector_type(4))) int          i32x4;

#if defined(__has_builtin)
#if __has_builtin(__builtin_amdgcn_tensor_load_to_lds) && \
    __has_builtin(__builtin_amdgcn_s_wait_tensorcnt)
#define USE_TDM 1
#endif
#endif

__device__ __forceinline__ void atomic_add_f32(float* p, float v) {
  (void)__hip_atomic_fetch_add(p, v, __ATOMIC_RELAXED, __HIP_MEMORY_SCOPE_AGENT);
}

#if USE_TDM
// Issue one TDM gather: 8 rows of h (each DD floats) -> LDS at lds_byte_addr.
// D# bit layout per CDNA5 ISA (gather mode,
// 32-bit indices, data_size=4B, tile_dim0=tensor_dim0=stride=128).
__device__ __forceinline__ void tdm_gather8(unsigned lds_byte_addr,
                                            const float* __restrict__ h,
                                            int n_rows, int nval, int col_lane) {
  u32x4 g0;
  g0.x = 0xC0000001u;                               // count=1 | idx32 | gather_mode
  g0.y = lds_byte_addr;                             // lds_addr
  unsigned long long ga = (unsigned long long)(const void*)h;
  g0.z = (unsigned)ga;                              // global_addr[31:0]
  g0.w = ((unsigned)(ga >> 32) & 0x01FFFFFFu) | 0x80000000u; // addr[56:32] | type=2

  i32x8 g1;
  g1.s0 = (int)(2u << 16);                          // data_size = 4 bytes
  g1.s1 = (int)((unsigned)DD << 16);                // tensor_dim0 lo16 = 128
  g1.s2 = (int)(((unsigned)n_rows & 0xFFFFu) << 16);// dim0 hi16=0 | tensor_dim1 lo16
  g1.s3 = (int)(((unsigned)n_rows >> 16) | ((unsigned)DD << 16)); // dim1 hi16 | tile_dim0
  g1.s4 = nval;                                     // tile_dim1 = #valid indices
  g1.s5 = DD;                                       // tensor_dim0_stride lo32 = 128
  g1.s6 = 0;
  g1.s7 = 0;

  i32x4 g2, g3;                                     // row indices (32-bit mode)
  g2.x = __builtin_amdgcn_readlane(col_lane, 0);
  g2.y = __builtin_amdgcn_readlane(col_lane, 1);
  g2.z = __builtin_amdgcn_readlane(col_lane, 2);
  g2.w = __builtin_amdgcn_readlane(col_lane, 3);
  g3.x = __builtin_amdgcn_readlane(col_lane, 4);
  g3.y = __builtin_amdgcn_readlane(col_lane, 5);
  g3.z = __builtin_amdgcn_readlane(col_lane, 6);
  g3.w = __builtin_amdgcn_readlane(col_lane, 7);

  i32x8 g4 = {0, 0, 0, 0, 0, 0, 0, 0};              // unused trailing operand group

  // amdgpu-toolchain (clang-23 / therock-10.0) 6-arg form.
  __builtin_amdgcn_tensor_load_to_lds(g0, g1, g2, g3, g4, 0);
}
#endif

// COO SpMM: out[r] += val * h[c], one wave handles CHUNK edges per step.
__global__ __launch_bounds__(SPMM_BLOCK) void spmm_coo_kernel(
    const int* __restrict__ rows, const int* __restrict__ cols,
    const float* __restrict__ vals, const float* __restrict__ h,
    float* __restrict__ out, const int nnz, const int h_rows) {
  const int lane = threadIdx.x & 31;
#if USE_TDM
  const int wave = threadIdx.x >> 5;
  __shared__ float tile[SPMM_WAVES][2][CHUNK * DD];   // 32 KB / block
#endif
  const long gw = (long)blockIdx.x * SPMM_WAVES + (threadIdx.x >> 5);
  const long nw = (long)gridDim.x * SPMM_WAVES;
  const long nchunks = ((long)nnz + CHUNK - 1) / CHUNK;
  if (gw >= nchunks) return;

#if USE_TDM
  const unsigned lds0 = (unsigned)(size_t)&tile[wave][0][0]; // low 32b of LDS aperture addr
  const unsigned lds1 = (unsigned)(size_t)&tile[wave][1][0];
  {
    const long e0 = gw * CHUNK;
    const int nval = (int)(((long)nnz - e0) < CHUNK ? ((long)nnz - e0) : (long)CHUNK);
    const int c = (lane < nval) ? cols[e0 + lane] : 0;
    tdm_gather8(lds0, h, h_rows, nval, c);
  }
  int buf = 0;
#endif

  for (long ci = gw; ci < nchunks; ci += nw) {
    const long e0 = ci * CHUNK;
    const int nval = (int)(((long)nnz - e0) < CHUNK ? ((long)nnz - e0) : (long)CHUNK);
    const int r  = (lane < nval) ? rows[e0 + lane] : 0;
    const int vb = (lane < nval) ? __float_as_int(vals[e0 + lane]) : 0;

#if USE_TDM
    const long cn = ci + nw;
    if (cn < nchunks) {
      const long e1 = cn * CHUNK;
      const int nv1 = (int)(((long)nnz - e1) < CHUNK ? ((long)nnz - e1) : (long)CHUNK);
      const int c1 = (lane < nv1) ? cols[e1 + lane] : 0;
      // make sure our previous ds reads of the target buffer retired
      asm volatile("s_wait_dscnt 0" ::: "memory");
      tdm_gather8(buf ? lds0 : lds1, h, h_rows, nv1, c1);
      __builtin_amdgcn_s_wait_tensorcnt(1);   // current buffer's gather complete
    } else {
      __builtin_amdgcn_s_wait_tensorcnt(0);
    }
    asm volatile("" ::: "memory");            // keep LDS reads below the wait
    const float* __restrict__ src = &tile[wave][buf][0];
#else
    const int c = (lane < nval) ? cols[e0 + lane] : 0;
#endif

#pragma unroll
    for (int j = 0; j < CHUNK; ++j) {
      if (j >= nval) break;
      const int   rj = __builtin_amdgcn_readlane(r, j);
      const float vj = __int_as_float(__builtin_amdgcn_readlane(vb, j));
#if USE_TDM
      const float4 hv = *(const float4*)(src + j * DD + lane * 4);  // ds_load_b128
#else
      const int cj = __builtin_amdgcn_readlane(c, j);
      const float4 hv = *(const float4*)(h + (long)cj * DD + lane * 4);
#endif
      float* dst = out + (long)rj * DD + lane * 4;
      atomic_add_f32(dst + 0, hv.x * vj);
      atomic_add_f32(dst + 1, hv.y * vj);
      atomic_add_f32(dst + 2, hv.z * vj);
      atomic_add_f32(dst + 3, hv.w * vj);
    }
#if USE_TDM
    buf ^= 1;
#endif
  }
}

// ---- trivial streaming kernels (float4-vectorized) ----
__global__ void fill_zero_kernel(float4* __restrict__ p, long n4) {
  long i = (long)blockIdx.x * blockDim.x + threadIdx.x;
  const long s = (long)gridDim.x * blockDim.x;
  const float4 z = make_float4(0.f, 0.f, 0.f, 0.f);
  for (; i < n4; i += s) p[i] = z;
}
__global__ void copy_kernel(float4* __restrict__ d, const float4* __restrict__ s4, long n4) {
  long i = (long)blockIdx.x * blockDim.x + threadIdx.x;
  const long s = (long)gridDim.x * blockDim.x;
  for (; i < n4; i += s) d[i] = s4[i];
}
__global__ void add_kernel(float4* __restrict__ acc, const float4* __restrict__ s4, long n4) {
  long i = (long)blockIdx.x * blockDim.x + threadIdx.x;
  const long s = (long)gridDim.x * blockDim.x;
  for (; i < n4; i += s) {
    float4 a = acc[i], b = s4[i];
    a.x += b.x; a.y += b.y; a.z += b.z; a.w += b.w;
    acc[i] = a;
  }
}
__global__ void scale_kernel(float4* __restrict__ p, long n4, float sc) {
  long i = (long)blockIdx.x * blockDim.x + threadIdx.x;
  const long s = (long)gridDim.x * blockDim.x;
  for (; i < n4; i += s) {
    float4 a = p[i];
    a.x *= sc; a.y *= sc; a.z *= sc; a.w *= sc;
    p[i] = a;
  }
}

extern "C" void kernel_launch(void* const* d_in, const int* in_sizes, int n_in,
                              void* d_out, int out_size, void* d_ws, size_t ws_size,
                              hipStream_t stream) {
  (void)n_in; (void)out_size; (void)ws_size;
  const float* user_emb = (const float*)d_in[0];
  const float* item_emb = (const float*)d_in[1];
  const int*   u1r = (const int*)d_in[2];
  const int*   u1c = (const int*)d_in[3];
  const float* u1v = (const float*)d_in[4];
  const int*   u2r = (const int*)d_in[5];
  const int*   u2c = (const int*)d_in[6];
  const float* u2v = (const float*)d_in[7];
  const int*   ucr = (const int*)d_in[8];
  const int*   ucc = (const int*)d_in[9];
  const float* ucv = (const float*)d_in[10];
  const int*   i1r = (const int*)d_in[11];
  const int*   i1c = (const int*)d_in[12];
  const float* i1v = (const float*)d_in[13];
  const int*   i2r = (const int*)d_in[14];
  const int*   i2c = (const int*)d_in[15];
  const float* i2v = (const float*)d_in[16];
  const int*   icr = (const int*)d_in[17];
  const int*   icc = (const int*)d_in[18];
  const float* icv = (const float*)d_in[19];

  const int NU   = in_sizes[0] / DD;   // 100000
  const int NI   = in_sizes[1] / DD;   // 50000
  const int nnzU = in_sizes[2];        // 3.2M
  const int nnzI = in_sizes[11];       // 1.6M

  float* accU = (float*)d_out;
  float* accI = accU + (size_t)NU * DD;

  // ping-pong buffers in workspace (3 x NU x D floats max)
  float* H = (float*)d_ws;
  float* A = H + (size_t)NU * DD;
  float* B = A + (size_t)NU * DD;

  auto ewb = [](long n4) {
    long b = (n4 + 255) / 256;
    if (b < 1) b = 1;
    if (b > 4096) b = 4096;
    return (int)b;
  };
  auto run_spmm = [&](const int* r, const int* c, const float* v,
                      const float* hsrc, float* o, int nnz, int nrows) {
    const long n4 = (long)nrows * DD / 4;
    fill_zero_kernel<<<ewb(n4), 256, 0, stream>>>((float4*)o, n4);
    long chunks = ((long)nnz + CHUNK - 1) / CHUNK;
    long b = (chunks + SPMM_WAVES * 4 - 1) / (SPMM_WAVES * 4);
    if (b < 1) b = 1;
    if (b > 8192) b = 8192;
    spmm_coo_kernel<<<(int)b, SPMM_BLOCK, 0, stream>>>(r, c, v, hsrc, o, nnz, nrows);
  };
  auto run_branch = [&](const float* emb, int n,
                        const int* r2, const int* c2, const float* v2,
                        const int* r1, const int* c1, const float* v1,
                        const int* rc, const int* cc, const float* vc,
                        int nnz, float* acc) {
    const long n4 = (long)n * DD / 4;
    copy_kernel<<<ewb(n4), 256, 0, stream>>>((float4*)acc, (const float4*)emb, n4);
    const float* src = emb;
    for (int l = 0; l < 3; ++l) {
      run_spmm(r2, c2, v2, src, A, nnz, n);   // t = spmm(m2, h)
      run_spmm(r1, c1, v1, A,   B, nnz, n);   // t = spmm(m1, t)
      run_spmm(rc, cc, vc, B,   H, nnz, n);   // t = spmm(mcat, t)
      add_kernel<<<ewb(n4), 256, 0, stream>>>((float4*)acc, (const float4*)H, n4);
      src = H;
    }
    scale_kernel<<<ewb(n4), 256, 0, stream>>>((float4*)acc, n4, 0.25f);
  };

  run_branch(user_emb, NU, u2r, u2c, u2v, u1r, u1c, u1v, ucr, ucc, ucv, nnzU, accU);
  run_branch(item_emb, NI, i2r, i2c, i2v, i1r, i1c, i1v, icr, icc, icv, nnzI, accI);
}